// DynamicCNN_54022098649658
// MI455X (gfx1250) — compile-verified
//
#include <hip/hip_runtime.h>
#include <math.h>

typedef __attribute__((ext_vector_type(2))) float v2f;
typedef __attribute__((ext_vector_type(8))) float v8f;

#define HW   512
#define KF   50
#define KPAD 64
#define LDSW 514                      // 512 + 2-column halo
#define TEMP_INV (1.0f/(5.0f+1e-5f))

// ---------------------------------------------------------------------------
// Kernel 1: Attention2D -> aw_pad[b][64] (softmax weights, zero padded k>=50)
// One block per batch. Deterministic (no float atomics).
// ---------------------------------------------------------------------------
__global__ __launch_bounds__(256)
void attention_kernel(const float* __restrict__ x,
                      const float* __restrict__ fc1_w,
                      const float* __restrict__ fc2_w,
                      const float* __restrict__ fc2_b,
                      const float* __restrict__ ln_g,
                      const float* __restrict__ ln_b,
                      float* __restrict__ aw_pad) {
  const int b = blockIdx.x;
  const int t = threadIdx.x;
  __shared__ float h[18*18];          // zero-padded 16x16 hidden map
  __shared__ float S9[9];             // shifted-window sums per tap
  __shared__ float ash[KF];

  for (int e = t; e < 18*18; e += 256) h[e] = 0.0f;
  __syncthreads();

  // AdaptiveAvgPool 512x512 -> 16x16 : each thread averages one 32x32 cell
  const int ci = t >> 4, cj = t & 15;
  const float* xb = x + (size_t)b * HW * HW;
  float s = 0.0f;
  for (int r = 0; r < 32; ++r) {
    const float* row = xb + (size_t)(ci*32 + r) * HW + cj*32;
    #pragma unroll
    for (int c = 0; c < 32; c += 4) {
      float4 v = *reinterpret_cast<const float4*>(row + c);
      s += v.x + v.y + v.z + v.w;
    }
  }
  float ctx = s * (1.0f/1024.0f);
  h[(ci+1)*18 + (cj+1)] = fmaxf(fc1_w[0]*ctx, 0.0f);   // fc1 (1x1) + ReLU
  __syncthreads();

  // mean over cells of (3x3 conv) == conv weights dot shifted window sums
  if (t < 9) {
    int dy = t/3, dx = t%3;
    float acc = 0.0f;
    for (int i = 0; i < 16; ++i)
      for (int j = 0; j < 16; ++j)
        acc += h[(i+dy)*18 + (j+dx)];
    S9[t] = acc;
  }
  __syncthreads();
  if (t < KF) {
    float a = fc2_b[t];
    #pragma unroll
    for (int tap = 0; tap < 9; ++tap)
      a += fc2_w[t*9+tap] * (S9[tap] * (1.0f/256.0f));
    ash[t] = a;
  }
  __syncthreads();

  if (t == 0) {                        // K=50: serial LN + softmax is free
    float mu = 0.0f;
    for (int k = 0; k < KF; ++k) mu += ash[k];
    mu *= (1.0f/KF);
    float var = 0.0f;
    for (int k = 0; k < KF; ++k) { float d = ash[k]-mu; var += d*d; }
    var *= (1.0f/KF);
    float inv = rsqrtf(var + 1e-5f);
    float z[KF], zmax = -1e30f;
    for (int k = 0; k < KF; ++k) {
      float an = (ash[k]-mu)*inv*ln_g[k] + ln_b[k];
      z[k] = an * TEMP_INV;
      zmax = fmaxf(zmax, z[k]);
    }
    float se = 0.0f;
    for (int k = 0; k < KF; ++k) { z[k] = __expf(z[k]-zmax); se += z[k]; }
    float rse = 1.0f/se;
    for (int k = 0;  k < KF;   ++k) aw_pad[b*KPAD + k] = z[k]*rse;
    for (int k = KF; k < KPAD; ++k) aw_pad[b*KPAD + k] = 0.0f;
  }
}

// ---------------------------------------------------------------------------
// Kernel 2: dynamic 3x3 conv (K=50 filters) + bias + sigmoid(out*aw)
// GEMM view: D(16k x 16px) = A(16x4 taps) x B(4x16) via V_WMMA_F32_16X16X4_F32
// One block per (b, y) row; 8 waves x 64 px each; x row strip staged in LDS.
// Row 3 of the LDS tile is a permanent zero row: invalid taps (t>=9) read it,
// keeping every B-fragment ds_load unconditional (no EXEC masking near WMMA).
// ---------------------------------------------------------------------------
__global__ __launch_bounds__(256)
void dynconv_kernel(const float* __restrict__ x,
                    const float* __restrict__ weight,
                    const float* __restrict__ bias,
                    const float* __restrict__ aw_pad,
                    float* __restrict__ out) {
  const int y    = blockIdx.x;
  const int b    = blockIdx.y;
  const int t    = threadIdx.x;
  const int lane = t & 31;
  const int wv   = t >> 5;
  const int n    = lane & 15;          // pixel column within 16-wide tile
  const int hi   = lane >> 4;          // half-wave selects tap pair / k offset

  __shared__ float tile[4*LDSW];       // rows y-1, y, y+1, zero-row

  const float* xb = x + (size_t)b * HW * HW;
  for (int e = t; e < 3*LDSW; e += 256) {
    int r = e / LDSW, c = e % LDSW;
    int gy = y + r - 1, gx = c - 1;
    float v = 0.0f;
    if (gy >= 0 && gy < HW && gx >= 0 && gx < HW) v = xb[(size_t)gy*HW + gx];
    tile[e] = v;
  }
  for (int e = t; e < LDSW; e += 256) tile[3*LDSW + e] = 0.0f;  // zero row

  // A fragments: afr[m][i], 16x4 f32 layout (lanes 0-15: taps 4i,4i+1;
  // lanes 16-31: taps 4i+2,4i+3). Taps >=9 and k>=50 zero-padded.
  v2f afr[4][3];
  #pragma unroll
  for (int m = 0; m < 4; ++m) {
    int k = m*16 + n;
    #pragma unroll
    for (int i = 0; i < 3; ++i) {
      int t0 = 4*i + 2*hi, t1 = t0 + 1;
      float a0 = (k < KF && t0 < 9) ? weight[k*9 + t0] : 0.0f;
      float a1 = (k < KF && t1 < 9) ? weight[k*9 + t1] : 0.0f;
      v2f av = {a0, a1};
      afr[m][i] = av;
    }
  }
  // Per-accumulator-slot attention weight & bias (k = m*16 + v + 8*hi)
  float awk[4][8], bik[4][8];
  #pragma unroll
  for (int m = 0; m < 4; ++m)
    #pragma unroll
    for (int v = 0; v < 8; ++v) {
      int k = m*16 + v + hi*8;
      awk[m][v] = aw_pad[b*KPAD + k];
      bik[m][v] = (k < KF) ? bias[k] : 0.0f;
    }

  // Per-lane LDS base offsets for the 6 tap loads (constant across groups)
  int boff[3][2];
  #pragma unroll
  for (int i = 0; i < 3; ++i) {
    #pragma unroll
    for (int j = 0; j < 2; ++j) {
      int tp = 4*i + 2*hi + j;
      int rr = (tp < 9) ? (tp/3) : 3;       // zero row for padded taps
      int dx = (tp < 9) ? (tp%3) : 0;
      boff[i][j] = rr*LDSW + n + dx;
    }
  }

  // Per-lane output base: out[((b*KF)*HW + y)*HW + n], k/px added per slot
  float* obase = out + ((size_t)(b*KF)*HW + (size_t)y)*HW + n;

  __syncthreads();

  #pragma unroll
  for (int g = 0; g < 4; ++g) {
    const int px0 = wv*64 + g*16;

    // B fragments from LDS: unconditional loads (zero row handles padding)
    v2f bfr[3];
    #pragma unroll
    for (int i = 0; i < 3; ++i) {
      float b0 = tile[boff[i][0] + px0];
      float b1 = tile[boff[i][1] + px0];
      v2f bv = {b0, b1};
      bfr[i] = bv;
    }

    v8f acc[4];
    #pragma unroll
    for (int m = 0; m < 4; ++m) {
      v8f c = {};
      c = __builtin_amdgcn_wmma_f32_16x16x4_f32(false, afr[m][0], false, bfr[0],
                                                (short)0, c, false, false);
      c = __builtin_amdgcn_wmma_f32_16x16x4_f32(false, afr[m][1], false, bfr[1],
                                                (short)0, c, false, false);
      c = __builtin_amdgcn_wmma_f32_16x16x4_f32(false, afr[m][2], false, bfr[2],
                                                (short)0, c, false, false);
      acc[m] = c;
    }

    // Fused epilogue: sigmoid((conv + bias_k) * aw_k); fast v_rcp_f32 recip.
    #pragma unroll
    for (int m = 0; m < 4; ++m) {
      #pragma unroll
      for (int v = 0; v < 8; ++v) {
        int k = m*16 + v + hi*8;
        if (k < KF) {
          float zv = (acc[m][v] + bik[m][v]) * awk[m][v];
          float r  = __builtin_amdgcn_rcpf(1.0f + __expf(-zv));
          obase[(size_t)k*HW*HW + px0] = r;
        }
      }
    }
  }
}

// ---------------------------------------------------------------------------
extern "C" void kernel_launch(void* const* d_in, const int* in_sizes, int n_in,
                              void* d_out, int out_size, void* d_ws, size_t ws_size,
                              hipStream_t stream) {
  (void)in_sizes; (void)n_in; (void)out_size; (void)ws_size;
  const float* x      = (const float*)d_in[0];
  const float* fc1_w  = (const float*)d_in[1];
  const float* fc2_w  = (const float*)d_in[2];
  const float* fc2_b  = (const float*)d_in[3];
  const float* ln_g   = (const float*)d_in[4];
  const float* ln_b   = (const float*)d_in[5];
  const float* weight = (const float*)d_in[6];
  const float* bias   = (const float*)d_in[7];
  float* out    = (float*)d_out;
  float* aw_pad = (float*)d_ws;       // 8*64 floats of scratch

  attention_kernel<<<8, 256, 0, stream>>>(x, fc1_w, fc2_w, fc2_b,
                                          ln_g, ln_b, aw_pad);
  dim3 grid(HW, 8);
  dynconv_kernel<<<grid, 256, 0, stream>>>(x, weight, bias, aw_pad, out);
}